// GATGraphConvLayer_12240656794082
// MI455X (gfx1250) — compile-verified
//
#include <hip/hip_runtime.h>
#include <math.h>

#define DFEAT 256
#define NEG_SLOPE 0.2f

typedef float v2f __attribute__((ext_vector_type(2)));
typedef float v8f __attribute__((ext_vector_type(8)));

// ---------------- init kernels ----------------

__global__ void k_zero(float* __restrict__ p, int n) {
  int i = blockIdx.x * blockDim.x + threadIdx.x;
  if (i < n) p[i] = 0.0f;
}

// zero rst[n_elems]; mmap = mapped(-FLT_MAX); s = 0 for n_nodes
__global__ void k_init_rel(float* __restrict__ rst, unsigned int* __restrict__ mmap,
                           float* __restrict__ s, int n_nodes, int n_elems) {
  int i = blockIdx.x * blockDim.x + threadIdx.x;
  if (i < n_elems) rst[i] = 0.0f;
  if (i < n_nodes) { mmap[i] = 0x00800000u; s[i] = 0.0f; }  // map(-FLT_MAX)
}

// ---------------- projection GEMM: feat = x @ Wr  (f32 WMMA 16x16x4) ----------------
// One wave computes one 16x16 tile of feat. K=256 in steps of 4.
// A (16x4 f32): v0 = K0(lanes0-15)/K2(lanes16-31), v1 = K1/K3  -> float2 per lane.
// B (4x16 f32): v0 = row K0/K2 striped across lanes, v1 = row K1/K3.
__global__ void k_gemm_feat(const float* __restrict__ x, const float* __restrict__ Wr,
                            float* __restrict__ feat, int nTiles) {
  const int lane = threadIdx.x & 31;
  const int wave = threadIdx.x >> 5;
  const int tile = blockIdx.x * 8 + wave;
  if (tile >= nTiles) return;                 // wave-uniform; EXEC stays all-1s
  const int rowTile = tile >> 4;              // D/16 == 16 column tiles
  const int colTile = tile & 15;
  const int row0 = rowTile * 16;
  const int col0 = colTile * 16;
  const int r  = lane & 15;
  const int hi = lane >> 4;                   // 0: K+0/1, 1: K+2/3
  const float* __restrict__ xrow = x  + (size_t)(row0 + r) * DFEAT;
  const float* __restrict__ wcol = Wr + col0 + r;
  v8f acc = {};
#pragma unroll 16
  for (int k = 0; k < DFEAT; k += 4) {
    v2f a = *(const v2f*)(xrow + k + 2 * hi);          // 8B aligned
    v2f b;
    b.x = wcol[(size_t)(k + 2 * hi)     * DFEAT];
    b.y = wcol[(size_t)(k + 2 * hi + 1) * DFEAT];
    acc = __builtin_amdgcn_wmma_f32_16x16x4_f32(
        /*neg_a=*/false, a, /*neg_b=*/false, b,
        /*c_mod=*/(short)0, acc, /*reuse_a=*/false, /*reuse_b=*/false);
  }
  // C layout: VGPR j -> row j (lanes 0-15) / row j+8 (lanes 16-31), col = lane&15
  float* __restrict__ orow = feat + (size_t)row0 * DFEAT + col0 + r;
#pragma unroll
  for (int j = 0; j < 8; ++j)
    orow[(size_t)(j + 8 * hi) * DFEAT] = acc[j];
}

// ---------------- attention logits: el = feat@al, er = feat@ar (wave per node) ----------------

__global__ void k_attn(const float* __restrict__ feat, const float* __restrict__ al,
                       const float* __restrict__ ar, float* __restrict__ el,
                       float* __restrict__ er, int n_nodes) {
  int node = (blockIdx.x * blockDim.x + threadIdx.x) >> 5;
  int lane = threadIdx.x & 31;
  if (node >= n_nodes) return;
  const float* __restrict__ f = feat + (size_t)node * DFEAT;
  float sl = 0.f, sr = 0.f;
#pragma unroll
  for (int j = 0; j < DFEAT / 32; ++j) {
    float v = f[lane + 32 * j];
    sl += v * al[lane + 32 * j];
    sr += v * ar[lane + 32 * j];
  }
#pragma unroll
  for (int o = 16; o > 0; o >>= 1) {
    sl += __shfl_down(sl, o, 32);
    sr += __shfl_down(sr, o, 32);
  }
  if (lane == 0) { el[node] = sl; er[node] = sr; }
}

// ---------------- edge pass 1: score + segment max (ordered-int mapping) ----------------

__global__ void k_edge_score(const int* __restrict__ src, const int* __restrict__ dst,
                             const float* __restrict__ el, const float* __restrict__ er,
                             float* __restrict__ ebuf, unsigned int* __restrict__ mmap, int nE) {
  int e = blockIdx.x * blockDim.x + threadIdx.x;
  if (e >= nE) return;
  float v = el[src[e]] + er[dst[e]];
  v = (v >= 0.f) ? v : v * NEG_SLOPE;         // leaky_relu
  ebuf[e] = v;
  unsigned int bits = __float_as_uint(v);
  unsigned int key  = (bits & 0x80000000u) ? ~bits : (bits | 0x80000000u); // monotone map
  atomicMax(mmap + dst[e], key);
}

// ---------------- edge pass 2: exp(e - max) + segment sum ----------------

__global__ void k_edge_exp(const int* __restrict__ dst, const unsigned int* __restrict__ mmap,
                           float* __restrict__ ebuf, float* __restrict__ s, int nE) {
  int e = blockIdx.x * blockDim.x + threadIdx.x;
  if (e >= nE) return;
  unsigned int key  = mmap[dst[e]];
  unsigned int bits = (key & 0x80000000u) ? (key & 0x7FFFFFFFu) : ~key;   // unmap
  float m  = __uint_as_float(bits);
  float ex = expf(ebuf[e] - m);
  ebuf[e]  = ex;
  atomicAdd(s + dst[e], ex);
}

// ---------------- edge pass 3: rst[dst] += alpha * feat[src] (wave per edge) ----------------

__global__ void k_edge_scatter(const int* __restrict__ src, const int* __restrict__ dst,
                               const float* __restrict__ ebuf, const float* __restrict__ s,
                               const float* __restrict__ feat, float* __restrict__ rst, int nE) {
  int e    = (blockIdx.x * blockDim.x + threadIdx.x) >> 5;
  int lane = threadIdx.x & 31;
  if (e >= nE) return;
  int sn = src[e], dn = dst[e];
  float alpha = ebuf[e] / s[dn];
  const float* __restrict__ f = feat + (size_t)sn * DFEAT;
  float* __restrict__ o       = rst  + (size_t)dn * DFEAT;
#pragma unroll
  for (int j = 0; j < DFEAT / 32; ++j)
    atomicAdd(o + lane + 32 * j, f[lane + 32 * j] * alpha);
}

// ---------------- per-relation epilogue: out += elu(rst + x) ----------------

__global__ void k_accum(const float* __restrict__ rst, const float* __restrict__ x,
                        float* __restrict__ out, int n) {
  int i = blockIdx.x * blockDim.x + threadIdx.x;
  if (i >= n) return;
  float v = rst[i] + x[i];
  v = (v > 0.f) ? v : expm1f(v);              // elu, alpha=1
  out[i] += v;
}

// ---------------- host ----------------

static inline int cdiv(int a, int b) { return (a + b - 1) / b; }

extern "C" void kernel_launch(void* const* d_in, const int* in_sizes, int n_in,
                              void* d_out, int out_size, void* d_ws, size_t ws_size,
                              hipStream_t stream) {
  const float* x      = (const float*)d_in[0];
  const float* W      = (const float*)d_in[1];
  const float* attn_l = (const float*)d_in[2];
  const float* attn_r = (const float*)d_in[3];
  const int*   src    = (const int*)d_in[4];
  const int*   dst    = (const int*)d_in[5];
  float*       out    = (float*)d_out;

  const int D = DFEAT;
  const int N = in_sizes[0] / D;       // 20000
  const int R = in_sizes[2] / D;       // 3
  const int E = in_sizes[4] / R;       // 160000
  const int ND = N * D;

  // workspace carve-up (~42 MB; fits easily in L2's 192 MB working set)
  char* ws = (char*)d_ws;
  size_t off = 0;
  auto carve = [&](size_t bytes) -> void* {
    void* p = ws + off;
    off += (bytes + 255) & ~(size_t)255;
    return p;
  };
  float*        feat = (float*)carve((size_t)ND * 4);
  float*        rst  = (float*)carve((size_t)ND * 4);
  float*        el   = (float*)carve((size_t)N * 4);
  float*        er   = (float*)carve((size_t)N * 4);
  unsigned int* mmap = (unsigned int*)carve((size_t)N * 4);
  float*        s    = (float*)carve((size_t)N * 4);
  float*        ebuf = (float*)carve((size_t)E * 4);
  (void)ws_size; (void)n_in; (void)out_size;

  const int nTiles = (N / 16) * (D / 16);   // 20000 (N divisible by 16)

  k_zero<<<cdiv(ND, 256), 256, 0, stream>>>(out, ND);

  for (int r = 0; r < R; ++r) {
    const float* Wr = W + (size_t)r * D * D;
    const float* al = attn_l + (size_t)r * D;
    const float* ar = attn_r + (size_t)r * D;
    const int*   sr = src + (size_t)r * E;
    const int*   dr = dst + (size_t)r * E;

    k_gemm_feat<<<cdiv(nTiles, 8), 256, 0, stream>>>(x, Wr, feat, nTiles);
    k_attn<<<cdiv(N, 8), 256, 0, stream>>>(feat, al, ar, el, er, N);
    k_init_rel<<<cdiv(ND, 256), 256, 0, stream>>>(rst, mmap, s, N, ND);
    k_edge_score<<<cdiv(E, 256), 256, 0, stream>>>(sr, dr, el, er, ebuf, mmap, E);
    k_edge_exp<<<cdiv(E, 256), 256, 0, stream>>>(dr, mmap, ebuf, s, E);
    k_edge_scatter<<<cdiv(E, 8), 256, 0, stream>>>(sr, dr, ebuf, s, feat, rst, E);
    k_accum<<<cdiv(ND, 256), 256, 0, stream>>>(rst, x, out, ND);
  }
}